// Classifier_46231027974392
// MI455X (gfx1250) — compile-verified
//
#include <hip/hip_runtime.h>
#include <hip/hip_bf16.h>
#include <math.h>

// ---------------------------------------------------------------------------
// Problem constants (from reference)
// ---------------------------------------------------------------------------
#define NN        50000      // nodes
#define E_RAW     800000     // edges without self loops
#define EP        850000     // edges + self loops
#define DIN       128
#define HC        128        // H*C
#define CC        64         // per-head channels == layer out dim
#define NCLS      4
#define EPS_GN    1e-5f

typedef float v2f __attribute__((ext_vector_type(2)));
typedef float v8f __attribute__((ext_vector_type(8)));

// ---------------------------------------------------------------------------
// Workspace layout (float offsets). Total ~22.0M floats (~88 MB).
// ---------------------------------------------------------------------------
#define WS_XL     0u          // [NN*HC]  x @ Wl + bl
#define WS_XR     6400000u    // [NN*HC]  x @ Wr + br
#define WS_HPRE   12800000u   // [NN*CC]  head-mean + bias (pre GraphNorm)
#define WS_HN     16000000u   // [NN*CC]  GraphNorm output (layer input / final h)
#define WS_CSRE   19200000u   // [EP*2]   per-(sorted edge, head) scores
#define WS_OFF    20900000u   // [NN+1]   (int) CSR row offsets (by dst)
#define WS_CUR    21000000u   // [NN]     (int) scatter cursors
#define WS_CSRS   21100000u   // [EP]     (int) source node per CSR slot
#define WS_STATS  22000000u   // [128]    colsum[64] | sqsum[64]

// ---------------------------------------------------------------------------
// Utility kernels
// ---------------------------------------------------------------------------
__global__ void fill_f32(float* __restrict__ p, float v, int n) {
    int i = blockIdx.x * blockDim.x + threadIdx.x;
    if (i < n) p[i] = v;
}

__device__ __forceinline__ float lrelu02(float v) {
    return v > 0.0f ? v : 0.2f * v;
}

__device__ __forceinline__ void edge_sd(const int* __restrict__ ei, int eid,
                                        int& s, int& d) {
    if (eid < E_RAW) { s = ei[eid]; d = ei[E_RAW + eid]; }
    else             { s = d = eid - E_RAW; }
}

// ---------------------------------------------------------------------------
// WMMA GEMM:  Y[NN, HC] = X[NN, K] @ W[K, HC] + bias[HC]
// One 256-thread block (8 waves) per 16-row tile; wave w handles column tile w.
// Uses V_WMMA_F32_16X16X4_F32 (full f32 precision, CDNA5 matrix pipe).
// ---------------------------------------------------------------------------
__global__ void __launch_bounds__(256) gemm_bias_wmma(
    const float* __restrict__ X, const float* __restrict__ W,
    const float* __restrict__ bias, float* __restrict__ Y, int K) {
    const int lane = threadIdx.x & 31;
    const int wave = threadIdx.x >> 5;       // column tile 0..7
    const int m0   = blockIdx.x * 16;
    const int n0   = wave * 16;
    const int g    = lane >> 4;              // half-wave select
    const int l15  = lane & 15;

    v8f acc = {};
    const float* Xrow = X + (size_t)(m0 + l15) * K;
    for (int k0 = 0; k0 < K; k0 += 4) {
        const int ka = k0 + g * 2;
        v2f a, b;
        a.x = Xrow[ka + 0];
        a.y = Xrow[ka + 1];
        b.x = W[(size_t)(ka + 0) * HC + n0 + l15];
        b.y = W[(size_t)(ka + 1) * HC + n0 + l15];
        acc = __builtin_amdgcn_wmma_f32_16x16x4_f32(
            false, a, false, b, (short)0, acc, false, false);
    }
    const int col = n0 + l15;
    const float bv = bias[col];
#pragma unroll
    for (int i = 0; i < 8; ++i) {
        const int row = m0 + i + g * 8;
        Y[(size_t)row * HC + col] = acc[i] + bv;
    }
}

// ---------------------------------------------------------------------------
// CSR build (by destination). Built once per call, reused by both layers.
// ---------------------------------------------------------------------------
__global__ void count_deg(const int* __restrict__ ei, int* __restrict__ off) {
    int eid = blockIdx.x * blockDim.x + threadIdx.x;
    if (eid >= EP) return;
    int s, d; edge_sd(ei, eid, s, d);
    atomicAdd(&off[d], 1);
}

#define SCAN_T  1024
#define CHUNK   ((NN + SCAN_T - 1) / SCAN_T)   // 49

// In-place exclusive scan of off[0..NN-1]; writes off[NN] = total.
__global__ void __launch_bounds__(SCAN_T) scan_offsets(int* __restrict__ off) {
    __shared__ int part[SCAN_T];
    const int t = threadIdx.x;
    const int b = t * CHUNK;
    const int e = (b + CHUNK < NN) ? (b + CHUNK) : NN;
    int s = 0;
    for (int i = b; i < e; ++i) s += off[i];
    part[t] = s;
    __syncthreads();
    // Hillis-Steele inclusive scan over 1024 partials
    for (int d = 1; d < SCAN_T; d <<= 1) {
        int v = (t >= d) ? part[t - d] : 0;
        __syncthreads();
        part[t] += v;
        __syncthreads();
    }
    int run = part[t] - s;   // exclusive base for this chunk
    for (int i = b; i < e; ++i) {
        int di = off[i];     // read before overwrite (thread-private element)
        off[i] = run;
        run += di;
    }
    if (t == SCAN_T - 1) off[NN] = part[SCAN_T - 1];
}

__global__ void copy_cursor(const int* __restrict__ off, int* __restrict__ cur) {
    int i = blockIdx.x * blockDim.x + threadIdx.x;
    if (i < NN) cur[i] = off[i];
}

__global__ void scatter_edges(const int* __restrict__ ei, int* __restrict__ cur,
                              int* __restrict__ csr_src) {
    int eid = blockIdx.x * blockDim.x + threadIdx.x;
    if (eid >= EP) return;
    int s, d; edge_sd(ei, eid, s, d);
    int pos = atomicAdd(&cur[d], 1);
    csr_src[pos] = s;
}

// ---------------------------------------------------------------------------
// GATv2 aggregation: one wave32 per node, in-edges walked sequentially.
// Lane j covers channels [4j, 4j+4) of the H*C=128 vector
// (lanes 0-15 -> head 0, lanes 16-31 -> head 1).
// sweep1: score per edge (shfl-xor half-wave dot), per-head max, store scores
// denom : coalesced read of stored scores, register exp-sum
// sweep2: alpha-weighted gather-accumulate of xl[src] in registers
// finish: head mean via shfl_xor(16), + bias, single b128 store of hpre row
// ---------------------------------------------------------------------------
__global__ void __launch_bounds__(256) node_aggregate(
    const int* __restrict__ off, const int* __restrict__ csr_src,
    const float* __restrict__ xl, const float* __restrict__ xr,
    const float* __restrict__ att, const float* __restrict__ bias,
    float* __restrict__ csr_e, float* __restrict__ hpre) {
    const int n = blockIdx.x * (blockDim.x >> 5) + (threadIdx.x >> 5);
    if (n >= NN) return;
    const int lane = threadIdx.x & 31;
    const int h    = lane >> 4;
    const int l15  = lane & 15;
    const int start = off[n];
    const int end   = off[n + 1];

    const float4 xrv = *(const float4*)(xr + (size_t)n * HC + lane * 4);
    const float4 av  = *(const float4*)(att + lane * 4);

    // ---- sweep 1: scores + per-head max -----------------------------------
    float mx = -INFINITY;
    for (int pos = start; pos < end; ++pos) {
        const int s = csr_src[pos];
        const float4 xlv = *(const float4*)(xl + (size_t)s * HC + lane * 4);
        float p = lrelu02(xlv.x + xrv.x) * av.x
                + lrelu02(xlv.y + xrv.y) * av.y
                + lrelu02(xlv.z + xrv.z) * av.z
                + lrelu02(xlv.w + xrv.w) * av.w;
        p += __shfl_xor(p, 1);
        p += __shfl_xor(p, 2);
        p += __shfl_xor(p, 4);
        p += __shfl_xor(p, 8);           // now uniform within each 16-lane half
        mx = fmaxf(mx, p);
        if (l15 == 0) csr_e[(size_t)pos * 2 + h] = p;
    }

    // ---- denom: sum of exp(score - max) per head --------------------------
    float den = 0.0f;
    for (int pos = start + l15; pos < end; pos += 16)
        den += expf(csr_e[(size_t)pos * 2 + h] - mx);
    den += __shfl_xor(den, 1);
    den += __shfl_xor(den, 2);
    den += __shfl_xor(den, 4);
    den += __shfl_xor(den, 8);
    const float invden = 1.0f / (den + 1e-16f);

    // ---- sweep 2: alpha-weighted accumulation in registers ----------------
    float4 o = make_float4(0.0f, 0.0f, 0.0f, 0.0f);
    for (int pos = start; pos < end; ++pos) {
        const int s = csr_src[pos];
        const float alpha = expf(csr_e[(size_t)pos * 2 + h] - mx) * invden;
        const float4 xlv = *(const float4*)(xl + (size_t)s * HC + lane * 4);
        o.x += alpha * xlv.x;
        o.y += alpha * xlv.y;
        o.z += alpha * xlv.z;
        o.w += alpha * xlv.w;
    }

    // ---- head mean + bias -> hpre[n, 0:64] --------------------------------
    float4 q;
    q.x = __shfl_xor(o.x, 16);
    q.y = __shfl_xor(o.y, 16);
    q.z = __shfl_xor(o.z, 16);
    q.w = __shfl_xor(o.w, 16);
    if (lane < 16) {
        const float4 bv = *(const float4*)(bias + lane * 4);
        float4 r;
        r.x = 0.5f * (o.x + q.x) + bv.x;
        r.y = 0.5f * (o.y + q.y) + bv.y;
        r.z = 0.5f * (o.z + q.z) + bv.z;
        r.w = 0.5f * (o.w + q.w) + bv.w;
        *(float4*)(hpre + (size_t)n * CC + lane * 4) = r;
    }
}

// ---------------------------------------------------------------------------
// GraphNorm: per-feature mean / var over all nodes, then normalize.
// Block = 64 features x 4 nodes; LDS partial reduce + 64 atomics per block.
// ---------------------------------------------------------------------------
__global__ void __launch_bounds__(256) node_colsum(
    const float* __restrict__ hpre, float* __restrict__ colsum) {
    const int c = threadIdx.x & 63;
    const int r = threadIdx.x >> 6;
    const int n = blockIdx.x * 4 + r;
    float v = (n < NN) ? hpre[(size_t)n * CC + c] : 0.0f;
    __shared__ float red[256];
    red[threadIdx.x] = v;
    __syncthreads();
    if (r == 0) {
        float s = red[c] + red[64 + c] + red[128 + c] + red[192 + c];
        atomicAdd(&colsum[c], s);
    }
}

__global__ void __launch_bounds__(256) node_varsum(
    const float* __restrict__ hpre, const float* __restrict__ colsum,
    const float* __restrict__ gna, float* __restrict__ sqsum) {
    const int c = threadIdx.x & 63;
    const int r = threadIdx.x >> 6;
    const int n = blockIdx.x * 4 + r;
    const float mean = colsum[c] * (1.0f / (float)NN);
    float v = 0.0f;
    if (n < NN) {
        const float xc = hpre[(size_t)n * CC + c] - gna[c] * mean;
        v = xc * xc;
    }
    __shared__ float red[256];
    red[threadIdx.x] = v;
    __syncthreads();
    if (r == 0) {
        float s = red[c] + red[64 + c] + red[128 + c] + red[192 + c];
        atomicAdd(&sqsum[c], s);
    }
}

__global__ void node_norm(
    const float* __restrict__ hpre, const float* __restrict__ colsum,
    const float* __restrict__ sqsum, const float* __restrict__ gna,
    const float* __restrict__ gng, const float* __restrict__ gnb,
    float* __restrict__ hout) {
    const int idx = blockIdx.x * blockDim.x + threadIdx.x;
    if (idx >= NN * CC) return;
    const int c = idx & 63;
    const float mean = colsum[c] * (1.0f / (float)NN);
    const float var  = sqsum[c] * (1.0f / (float)NN);
    const float xc   = hpre[idx] - gna[c] * mean;
    hout[idx] = gng[c] * xc * rsqrtf(var + EPS_GN) + gnb[c];
}

// ---------------------------------------------------------------------------
// Classifier + log_softmax (4 classes). One thread per node.
// ---------------------------------------------------------------------------
__global__ void classify_logsoftmax(
    const float* __restrict__ h, const float* __restrict__ W1,
    const float* __restrict__ b1, float* __restrict__ out) {
    const int n = blockIdx.x * blockDim.x + threadIdx.x;
    if (n >= NN) return;
    float l0 = b1[0], l1 = b1[1], l2 = b1[2], l3 = b1[3];
    const float* hr = h + (size_t)n * CC;
#pragma unroll 4
    for (int c = 0; c < CC; ++c) {
        const float x = hr[c];
        l0 += x * W1[c * NCLS + 0];
        l1 += x * W1[c * NCLS + 1];
        l2 += x * W1[c * NCLS + 2];
        l3 += x * W1[c * NCLS + 3];
    }
    float m = fmaxf(fmaxf(l0, l1), fmaxf(l2, l3));
    float sum = expf(l0 - m) + expf(l1 - m) + expf(l2 - m) + expf(l3 - m);
    float lse = m + logf(sum);
    out[(size_t)n * NCLS + 0] = l0 - lse;
    out[(size_t)n * NCLS + 1] = l1 - lse;
    out[(size_t)n * NCLS + 2] = l2 - lse;
    out[(size_t)n * NCLS + 3] = l3 - lse;
}

// ---------------------------------------------------------------------------
// Host-side orchestration
// ---------------------------------------------------------------------------
static void run_gat_layer(const float* X, int K,
                          const float* Wl, const float* bl,
                          const float* Wr, const float* br,
                          const float* att, const float* bias,
                          const float* gng, const float* gnb, const float* gna,
                          float* ws, hipStream_t stream) {
    float* XL    = ws + WS_XL;
    float* XR    = ws + WS_XR;
    float* HPRE  = ws + WS_HPRE;
    float* HN    = ws + WS_HN;
    float* CSRE  = ws + WS_CSRE;
    int*   OFF   = (int*)(ws + WS_OFF);
    int*   CSRS  = (int*)(ws + WS_CSRS);
    float* STATS = ws + WS_STATS;   // [0,64) colsum, [64,128) sqsum

    // GEMMs: XL = X@Wl + bl, XR = X@Wr + br   (WMMA f32)
    gemm_bias_wmma<<<NN / 16, 256, 0, stream>>>(X, Wl, bl, XL, K);
    gemm_bias_wmma<<<NN / 16, 256, 0, stream>>>(X, Wr, br, XR, K);

    fill_f32<<<1, 128, 0, stream>>>(STATS, 0.0f, 128);

    // Per-node attention softmax + aggregation (no output atomics)
    node_aggregate<<<(NN + 7) / 8, 256, 0, stream>>>(
        OFF, CSRS, XL, XR, att, bias, CSRE, HPRE);

    // GraphNorm
    node_colsum<<<(NN + 3) / 4, 256, 0, stream>>>(HPRE, STATS);
    node_varsum<<<(NN + 3) / 4, 256, 0, stream>>>(HPRE, STATS, gna, STATS + 64);
    node_norm<<<(NN * CC + 255) / 256, 256, 0, stream>>>(
        HPRE, STATS, STATS + 64, gna, gng, gnb, HN);
}

extern "C" void kernel_launch(void* const* d_in, const int* in_sizes, int n_in,
                              void* d_out, int out_size, void* d_ws, size_t ws_size,
                              hipStream_t stream) {
    const float* x    = (const float*)d_in[0];
    const int*   ei   = (const int*)d_in[1];
    const float* Wl1  = (const float*)d_in[2];
    const float* bl1  = (const float*)d_in[3];
    const float* Wr1  = (const float*)d_in[4];
    const float* br1  = (const float*)d_in[5];
    const float* att1 = (const float*)d_in[6];
    const float* bias1= (const float*)d_in[7];
    const float* gng1 = (const float*)d_in[8];
    const float* gnb1 = (const float*)d_in[9];
    const float* gna1 = (const float*)d_in[10];
    const float* Wl2  = (const float*)d_in[11];
    const float* bl2  = (const float*)d_in[12];
    const float* Wr2  = (const float*)d_in[13];
    const float* br2  = (const float*)d_in[14];
    const float* att2 = (const float*)d_in[15];
    const float* bias2= (const float*)d_in[16];
    const float* gng2 = (const float*)d_in[17];
    const float* gnb2 = (const float*)d_in[18];
    const float* gna2 = (const float*)d_in[19];
    const float* W1   = (const float*)d_in[20];
    const float* b1   = (const float*)d_in[21];

    float* ws  = (float*)d_ws;
    float* HN  = ws + WS_HN;
    int*   OFF = (int*)(ws + WS_OFF);
    int*   CUR = (int*)(ws + WS_CUR);
    int*   CSRS= (int*)(ws + WS_CSRS);

    // ---- Build dst-CSR once (edge_index is layer-invariant) ---------------
    fill_f32<<<(NN + 1 + 255) / 256, 256, 0, stream>>>((float*)OFF, 0.0f, NN + 1);
    count_deg<<<(EP + 255) / 256, 256, 0, stream>>>(ei, OFF);
    scan_offsets<<<1, SCAN_T, 0, stream>>>(OFF);
    copy_cursor<<<(NN + 255) / 256, 256, 0, stream>>>(OFF, CUR);
    scatter_edges<<<(EP + 255) / 256, 256, 0, stream>>>(ei, CUR, CSRS);

    // ---- Layer 1: x [NN,128] -> HN [NN,64] --------------------------------
    run_gat_layer(x, DIN, Wl1, bl1, Wr1, br1, att1, bias1,
                  gng1, gnb1, gna1, ws, stream);
    // ---- Layer 2: HN [NN,64] -> HN [NN,64] (GEMMs read HN before node_norm
    //      overwrites it) ---------------------------------------------------
    run_gat_layer(HN, CC, Wl2, bl2, Wr2, br2, att2, bias2,
                  gng2, gnb2, gna2, ws, stream);

    // ---- Classifier + log_softmax -----------------------------------------
    classify_logsoftmax<<<(NN + 255) / 256, 256, 0, stream>>>(
        HN, W1, b1, (float*)d_out);
}